// Capsule_24249385353656
// MI455X (gfx1250) — compile-verified
//
#include <hip/hip_runtime.h>
#include <hip/hip_bf16.h>

// CDNA5 / gfx1250 capsule routing, algebraically fused.
// Precision: all WMMA are V_WMMA_F32_16X16X4_F32 (exact fp32).

typedef float v2f __attribute__((ext_vector_type(2)));
typedef float v8f __attribute__((ext_vector_type(8)));

#define CAPS_B   64
#define CAPS_N   8192
#define CAPS_D   64     // input dim per capsule vec
#define CAPS_K   16     // num_capsule
#define CAPS_d   16     // dim_capsule
#define WGS_PER_B 16
#define TILES_PER_WAVE 4   // 16 WGs * 8 waves * 4 tiles * 16 rows = 8192 = N

// ---------------------------------------------------------------------------
// Pass 0 helper: R[b,j] = sum_n u[b,n,j]
// ---------------------------------------------------------------------------
__global__ __launch_bounds__(64) void caps_rowsum(const float* __restrict__ U,
                                                  float* __restrict__ R) {
  int b  = blockIdx.x >> 6;
  int sl = blockIdx.x & 63;          // 64 slices of 128 rows
  int j  = threadIdx.x;              // 0..63
  const float* p = U + ((size_t)b * CAPS_N + (size_t)sl * 128) * CAPS_D + j;
  float acc = 0.f;
#pragma unroll 8
  for (int i = 0; i < 128; ++i) acc += p[(size_t)i * CAPS_D];
  atomicAdd(&R[b * CAPS_D + j], acc);
}

// ---------------------------------------------------------------------------
// Small per-iteration kernel: s -> squash -> (w accumulate) -> P (or d_out).
//  mode 0: G = R (B,64);  s[k,d] = (1/16) * sum_j R[j] W[j,16k+d]; w = v
//  mode 1: G = Q (B,16,64); s[k,d] = sum_j Q[k,j] W[j,16k+d];      w += v
//  mode 2: like 1, but write squash(s) to out and stop.
//  modes 0/1 then build P[b,j,k] = sum_d W[j,16k+d] * w[k,d].
// ---------------------------------------------------------------------------
__global__ __launch_bounds__(256) void caps_small(int mode,
                                                  const float* __restrict__ G,
                                                  const float* __restrict__ W,
                                                  float* __restrict__ wsum,
                                                  float* __restrict__ P,
                                                  float* __restrict__ out) {
  __shared__ float ws[CAPS_K * CAPS_d];
  int b = blockIdx.x;
  int t = threadIdx.x;           // t = k*16 + d
  int k = t >> 4;

  float s = 0.f;
  if (mode == 0) {
    const float* R = G + b * CAPS_D;
#pragma unroll 8
    for (int j = 0; j < CAPS_D; ++j) s += R[j] * W[j * 256 + t];
    s *= (1.0f / 16.0f);
  } else {
    const float* Qb = G + ((size_t)b * CAPS_K + k) * CAPS_D;
#pragma unroll 8
    for (int j = 0; j < CAPS_D; ++j) s += Qb[j] * W[j * 256 + t];
  }

  // squash: norm over d (16 adjacent lanes share k; wave32 shfl widths)
  float s2 = s * s;
#pragma unroll
  for (int m = 1; m < 16; m <<= 1) s2 += __shfl_xor(s2, m, 32);
  float scale = s2 / ((1.0f + s2) * sqrtf(s2 + 1e-7f));
  float v = scale * s;

  if (mode == 2) {
    out[b * (CAPS_K * CAPS_d) + t] = v;
    return;
  }
  float wv = (mode == 0) ? v : (wsum[b * 256 + t] + v);
  wsum[b * 256 + t] = wv;
  ws[t] = wv;
  __syncthreads();

  // P[b][j][k] : 1024 entries, 4 per thread
#pragma unroll
  for (int i = 0; i < 4; ++i) {
    int idx = t + 256 * i;
    int j = idx >> 4, kk = idx & 15;
    float acc = 0.f;
#pragma unroll
    for (int dd = 0; dd < 16; ++dd)
      acc += W[j * 256 + kk * 16 + dd] * ws[kk * 16 + dd];
    P[((size_t)b * CAPS_D + j) * CAPS_K + kk] = acc;
  }
}

// ---------------------------------------------------------------------------
// Big streaming pass: for each 16-row n-tile:
//   L(16n x 16k) = U(16x64) @ P(64x16)      -- 16x V_WMMA_F32_16X16X4_F32
//   c = softmax_k(L)
//   Q(16k x 64j) += c^T(16x16) @ U(16x64)   -- 16x V_WMMA_F32_16X16X4_F32
// Q accumulated per-wave in VGPRs, reduced via LDS ds_add_f32, then ~1 global
// atomic per element per workgroup.
// ---------------------------------------------------------------------------
__global__ __launch_bounds__(256) void caps_route_pass(
    const float* __restrict__ U,   // (B, N, 64)
    const float* __restrict__ P,   // (B, 64, 16)
    float* __restrict__ Q) {       // (B, 16, 64), pre-zeroed
  // padded LDS: row stride 68 floats -> float4-aligned + conflict-free column reads
  __shared__ float Ul[8][16][68];
  __shared__ float Cl[8][16][17];
  __shared__ float Qs[CAPS_K * CAPS_D];

  const int w    = threadIdx.x >> 5;
  const int lane = threadIdx.x & 31;
  const int half = lane >> 4;          // 0: lanes 0-15, 1: lanes 16-31
  const int lm   = lane & 15;
  const int b    = blockIdx.x / WGS_PER_B;
  const int wgi  = blockIdx.x % WGS_PER_B;

  for (int i = threadIdx.x; i < CAPS_K * CAPS_D; i += 256) Qs[i] = 0.f;

  // B-operand (P) for the logit GEMM, resident in VGPRs for the whole kernel.
  // f32 WMMA B layout: lanes 0-15 -> K rows {0,1}, lanes 16-31 -> {2,3}, N=lane%16
  v2f Pb[16];
  {
    const float* Pg = P + (size_t)b * CAPS_D * CAPS_K;
#pragma unroll
    for (int t = 0; t < 16; ++t) {
      int k0 = 4 * t + 2 * half;
      Pb[t].x = Pg[k0 * CAPS_K + lm];
      Pb[t].y = Pg[(k0 + 1) * CAPS_K + lm];
    }
  }

  v8f Qacc[4] = {};   // Q tile: M=k rows, 4 j-tiles of 16 cols

  __syncthreads();

  const int tile0 = wgi * (8 * TILES_PER_WAVE) + w * TILES_PER_WAVE;
  for (int it = 0; it < TILES_PER_WAVE; ++it) {
    const int n0 = (tile0 + it) * 16;
    const float* Ug = U + ((size_t)b * CAPS_N + n0) * CAPS_D;

    if (it + 1 < TILES_PER_WAVE) {
      // prefetch next 4KB tile -> global_prefetch_b8
      __builtin_prefetch(Ug + 16 * CAPS_D + lane * 32, 0, 1);
    }

    // ---- stage U tile (16x64 f32) into LDS, coalesced float4 ----
#pragma unroll
    for (int i = 0; i < 8; ++i) {
      int idx = lane + 32 * i;           // 256 float4 slots
      int r = idx >> 4, c4 = idx & 15;
      float4 v4 = ((const float4*)Ug)[r * 16 + c4];
      *(float4*)&Ul[w][r][c4 * 4] = v4;
    }
    __syncthreads();

    // ---- L = U @ P : M=n, N=k, K=64 in 16 steps of 4 ----
    v8f L = {};
#pragma unroll
    for (int t = 0; t < 16; ++t) {
      int col = 4 * t + 2 * half;        // A layout: K pair per lane-half
      v2f a;
      a.x = Ul[w][lm][col];
      a.y = Ul[w][lm][col + 1];
      L = __builtin_amdgcn_wmma_f32_16x16x4_f32(
          false, a, false, Pb[t], (short)0, L, false, false);
    }

    // ---- softmax over k (cols -> 16-lane groups), 8 independent rows ----
    float mx[8], ex[8], sm[8];
#pragma unroll
    for (int r = 0; r < 8; ++r) mx[r] = L[r];
#pragma unroll
    for (int m = 1; m < 16; m <<= 1)
#pragma unroll
      for (int r = 0; r < 8; ++r) mx[r] = fmaxf(mx[r], __shfl_xor(mx[r], m, 32));
#pragma unroll
    for (int r = 0; r < 8; ++r) { ex[r] = __expf(L[r] - mx[r]); sm[r] = ex[r]; }
#pragma unroll
    for (int m = 1; m < 16; m <<= 1)
#pragma unroll
      for (int r = 0; r < 8; ++r) sm[r] += __shfl_xor(sm[r], m, 32);

    // store c transposed: Cl[k][n]
#pragma unroll
    for (int r = 0; r < 8; ++r) Cl[w][lm][r + 8 * half] = ex[r] / sm[r];
    __syncthreads();

    // ---- Q += c^T @ U : M=k, N=j (4 tiles), K=n=16 in 4 steps of 4 ----
#pragma unroll
    for (int s4 = 0; s4 < 4; ++s4) {
      int nrow = 4 * s4 + 2 * half;
      v2f a;
      a.x = Cl[w][lm][nrow];
      a.y = Cl[w][lm][nrow + 1];
#pragma unroll
      for (int jt = 0; jt < 4; ++jt) {
        v2f bb;
        bb.x = Ul[w][nrow][jt * 16 + lm];
        bb.y = Ul[w][nrow + 1][jt * 16 + lm];
        Qacc[jt] = __builtin_amdgcn_wmma_f32_16x16x4_f32(
            false, a, false, bb, (short)0, Qacc[jt], false, false);
      }
    }
    __syncthreads();   // protect Ul/Cl before next tile overwrites them
  }

  // ---- reduce 8 waves' Q tiles in LDS (ds_add_f32), then global atomics ----
#pragma unroll
  for (int jt = 0; jt < 4; ++jt)
#pragma unroll
    for (int r = 0; r < 8; ++r) {
      int k = r + 8 * half;              // C/D layout: VGPR r -> M = r + 8*half
      atomicAdd(&Qs[k * CAPS_D + jt * 16 + lm], Qacc[jt][r]);
    }
  __syncthreads();

  float* Qb = Q + (size_t)b * CAPS_K * CAPS_D;
  for (int i = threadIdx.x; i < CAPS_K * CAPS_D; i += 256)
    atomicAdd(&Qb[i], Qs[i]);
}

// ---------------------------------------------------------------------------
extern "C" void kernel_launch(void* const* d_in, const int* in_sizes, int n_in,
                              void* d_out, int out_size, void* d_ws, size_t ws_size,
                              hipStream_t stream) {
  const float* u = (const float*)d_in[0];   // (64, 8192, 64) f32
  const float* W = (const float*)d_in[1];   // (64, 256) f32
  float* out = (float*)d_out;               // (64, 16, 16) f32

  float* ws   = (float*)d_ws;
  float* R    = ws;                 // 4096
  float* Q    = R + 4096;           // 65536
  float* wsum = Q + 65536;          // 16384
  float* P    = wsum + 16384;       // 65536  -> total 151552 floats (~592 KB)

  hipMemsetAsync(R, 0, 4096 * sizeof(float), stream);

  // iteration 0: uniform c -> row-sum suffices
  caps_rowsum<<<CAPS_B * 64, 64, 0, stream>>>(u, R);
  caps_small<<<CAPS_B, 256, 0, stream>>>(0, R, W, wsum, P, out);

  // iteration 1
  hipMemsetAsync(Q, 0, 65536 * sizeof(float), stream);
  caps_route_pass<<<CAPS_B * WGS_PER_B, 256, 0, stream>>>(u, P, Q);
  caps_small<<<CAPS_B, 256, 0, stream>>>(1, Q, W, wsum, P, out);

  // iteration 2 (final): output squash(s)
  hipMemsetAsync(Q, 0, 65536 * sizeof(float), stream);
  caps_route_pass<<<CAPS_B * WGS_PER_B, 256, 0, stream>>>(u, P, Q);
  caps_small<<<CAPS_B, 256, 0, stream>>>(2, Q, W, wsum, P, out);
}